// Attn_37417755083259
// MI455X (gfx1250) — compile-verified
//
#include <hip/hip_runtime.h>
#include <hip/hip_bf16.h>
#include <math.h>

typedef __attribute__((ext_vector_type(2))) float v2f;
typedef __attribute__((ext_vector_type(8))) float v8f;

#define HDIM 1024
#define BDIM 32
#define SDIM 4096

// ---------------------------------------------------------------------------
// Kernel 1: v[b,h] = sum_g hidden[b,g] * W[g,h]
// M=32 (batch), K=1024, N=1024. One wave per 16-wide N tile; each wave
// computes both 16-row M tiles so the B fragment is loaded once.
// Uses full-precision V_WMMA_F32_16X16X4_F32 (exact fp32, matches reference).
// ---------------------------------------------------------------------------
__global__ __launch_bounds__(32) void proj_wmma_kernel(
    const float* __restrict__ hidden,   // [32, 1024] row-major
    const float* __restrict__ W,        // [1024, 1024] row-major, W[g*H + h]
    float* __restrict__ v)              // [32, 1024]
{
    const int lane = threadIdx.x;        // full wave32, EXEC all ones
    const int n0   = blockIdx.x * 16;    // 64 blocks cover N=1024
    const int half = lane >> 4;          // 0: lanes 0-15, 1: lanes 16-31
    const int l16  = lane & 15;

    v8f acc0 = {};   // M rows 0..15
    v8f acc1 = {};   // M rows 16..31

    for (int k = 0; k < HDIM; k += 4) {
        const int krow = k + (half << 1);          // K=0,1 or K=2,3 per ISA layout
        // B fragment (4x16, KxN): one row of W per VGPR, striped across lanes
        v2f bf;
        bf.x = W[(size_t)(krow + 0) * HDIM + n0 + l16];
        bf.y = W[(size_t)(krow + 1) * HDIM + n0 + l16];
        // A fragments (16x4, MxK): lane holds 2 consecutive K values of row M
        v2f a0 = *(const v2f*)(hidden + (size_t)(l16)      * HDIM + krow);
        v2f a1 = *(const v2f*)(hidden + (size_t)(l16 + 16) * HDIM + krow);

        acc0 = __builtin_amdgcn_wmma_f32_16x16x4_f32(false, a0, false, bf,
                                                     (short)0, acc0, false, false);
        acc1 = __builtin_amdgcn_wmma_f32_16x16x4_f32(false, a1, false, bf,
                                                     (short)0, acc1, false, false);
    }

    // D layout (16x16 f32): VGPR j -> M=j (lanes 0-15) / M=j+8 (lanes 16-31)
#pragma unroll
    for (int j = 0; j < 8; ++j) {
        const int m = j + (half << 3);
        v[(size_t)(m)      * HDIM + n0 + l16] = acc0[j];
        v[(size_t)(m + 16) * HDIM + n0 + l16] = acc1[j];
    }
}

// ---------------------------------------------------------------------------
// Kernel 2 (bandwidth-bound pass): energies[b,s] = v[b,:] . enc[s,b,:]
// One wave per (b,s). Lanes stream the 4KB row with coalesced float4 loads;
// wave32 shuffle reduction. Blocks grouped by b so v[b] stays hot in cache.
// Streams encoder_outputs (512 MB) exactly once -> ~22 us at 23.3 TB/s.
// ---------------------------------------------------------------------------
__global__ __launch_bounds__(256) void energies_kernel(
    const float* __restrict__ enc,   // [S, B, H]
    const float* __restrict__ v,     // [B, H]
    float* __restrict__ energies)    // [B, S]
{
    const int wave = blockIdx.x * 8 + (threadIdx.x >> 5);
    const int lane = threadIdx.x & 31;
    const int b = wave >> 12;        // S = 4096 = 2^12
    const int s = wave & (SDIM - 1);

    const float4* __restrict__ erow = (const float4*)(enc + ((size_t)s * BDIM + b) * HDIM);
    const float4* __restrict__ vrow = (const float4*)(v + (size_t)b * HDIM);

    float acc = 0.0f;
#pragma unroll
    for (int i = lane; i < HDIM / 4; i += 32) {   // 8 iterations of b128 loads
        float4 e = erow[i];
        float4 x = vrow[i];
        acc += e.x * x.x + e.y * x.y + e.z * x.z + e.w * x.w;
    }
#pragma unroll
    for (int off = 16; off > 0; off >>= 1)
        acc += __shfl_down(acc, off, 32);

    if (lane == 0)
        energies[(size_t)b * SDIM + s] = acc;
}

// ---------------------------------------------------------------------------
// Kernel 3: row softmax over s (4096) for each b. Note: the bias term
// hidden[b].attn_b is constant per row and cancels under softmax, so it is
// (exactly) omitted.
// ---------------------------------------------------------------------------
__global__ __launch_bounds__(256) void softmax_kernel(
    const float* __restrict__ energies,  // [B, S]
    float* __restrict__ out)             // [B, 1, S]
{
    const int b   = blockIdx.x;
    const int tid = threadIdx.x;
    __shared__ float red[8];

    const float* __restrict__ row  = energies + (size_t)b * SDIM;
    float* __restrict__       orow = out      + (size_t)b * SDIM;

    // Pass 1: max
    float m = -INFINITY;
    for (int i = tid; i < SDIM; i += 256) m = fmaxf(m, row[i]);
#pragma unroll
    for (int off = 16; off > 0; off >>= 1) m = fmaxf(m, __shfl_down(m, off, 32));
    if ((tid & 31) == 0) red[tid >> 5] = m;
    __syncthreads();
    m = red[0];
#pragma unroll
    for (int i = 1; i < 8; ++i) m = fmaxf(m, red[i]);
    __syncthreads();

    // Pass 2: exp + sum (write exps so they aren't recomputed)
    float s = 0.0f;
    for (int i = tid; i < SDIM; i += 256) {
        float t = __expf(row[i] - m);
        orow[i] = t;
        s += t;
    }
#pragma unroll
    for (int off = 16; off > 0; off >>= 1) s += __shfl_down(s, off, 32);
    if ((tid & 31) == 0) red[tid >> 5] = s;
    __syncthreads();
    s = 0.0f;
#pragma unroll
    for (int i = 0; i < 8; ++i) s += red[i];

    // Pass 3: normalize
    const float inv = 1.0f / s;
    for (int i = tid; i < SDIM; i += 256) orow[i] *= inv;
}

// ---------------------------------------------------------------------------
extern "C" void kernel_launch(void* const* d_in, const int* in_sizes, int n_in,
                              void* d_out, int out_size, void* d_ws, size_t ws_size,
                              hipStream_t stream) {
    const float* hidden = (const float*)d_in[0];   // [1, 32, 1024]
    const float* enc    = (const float*)d_in[1];   // [4096, 32, 1024]
    const float* W      = (const float*)d_in[2];   // [1024, 1024]
    // d_in[3] = attn_b : mathematically irrelevant under row softmax (shift invariance)
    float* out = (float*)d_out;                    // [32, 1, 4096]

    float* v        = (float*)d_ws;                // 32*1024 floats = 128 KB
    float* energies = v + BDIM * HDIM;             // 32*4096 floats = 512 KB

    // v = hidden @ W via fp32 WMMA (64 waves, one per 16-wide N tile)
    proj_wmma_kernel<<<dim3(HDIM / 16), 32, 0, stream>>>(hidden, W, v);

    // energies[b,s] = v[b] . enc[s,b]  (one wave per (b,s), 8 waves/block)
    energies_kernel<<<dim3((BDIM * SDIM) / 8), 256, 0, stream>>>(enc, v, energies);

    // softmax over s per batch row
    softmax_kernel<<<dim3(BDIM), 256, 0, stream>>>(energies, out);
}